// GraphVToV_9955734192542
// MI455X (gfx1250) — compile-verified
//
#include <hip/hip_runtime.h>

typedef __attribute__((ext_vector_type(2))) float v2f;
typedef __attribute__((ext_vector_type(4))) float v4f;
typedef __attribute__((ext_vector_type(8))) float v8f;

// Problem constants (from reference): B=8, N=128, C=3, F_in=F_out=128.
// Rows = B*N*C = 3072. X:(3072,128), W:(256,128), bias:(128).
// L,R : (3072,128) f32 in workspace. out:(8,128,128,3,128).
#define ROWS   3072
#define FDIM   128
#define MTILES (ROWS / 16)   // 192
#define NTILES (FDIM / 16)   // 8

// ---------------------------------------------------------------------------
// Stage 1: L = X @ W1 + bias ; R = X @ W2   (f32 WMMA 16x16x4, one wave/tile)
// ---------------------------------------------------------------------------
__global__ void __launch_bounds__(128)
gemm_lr_kernel(const float* __restrict__ X,
               const float* __restrict__ W,
               const float* __restrict__ bias,
               float* __restrict__ L,
               float* __restrict__ R) {
  const int lane  = threadIdx.x & 31;
  const int gwave = blockIdx.x * (blockDim.x >> 5) + (threadIdx.x >> 5);

  // tile decomposition: gwave = which*NTILES*MTILES + ntile*MTILES + mtile
  const int mtile = gwave % MTILES;
  const int ntile = (gwave / MTILES) % NTILES;
  const int which = gwave / (MTILES * NTILES);   // 0 -> W1/L, 1 -> W2/R

  const int m0    = mtile * 16;
  const int n0    = ntile * 16;
  const int half  = lane >> 4;          // 0: K pair {0,1}; 1: K pair {2,3}
  const int mrow  = m0 + (lane & 15);   // A-matrix row held by this lane
  const int ncol  = n0 + (lane & 15);   // B/C/D column held by this lane
  const float* Wb = W + which * FDIM * FDIM;   // W1 or W2 (row-major k*128+n)

  v8f acc = {};
  #pragma unroll 4
  for (int k0 = 0; k0 < FDIM; k0 += 4) {
    const int kA = k0 + half * 2;
    // A frag: lane holds X[mrow][kA], X[mrow][kA+1]  (contiguous -> b64 load)
    v2f a = *(const v2f*)(X + mrow * FDIM + kA);
    // B frag: v0 = W[kA][ncol], v1 = W[kA+1][ncol]
    v2f b;
    b.x = Wb[kA * FDIM + ncol];
    b.y = Wb[(kA + 1) * FDIM + ncol];
    // 8 args: (neg_a, A, neg_b, B, c_mod, C, reuse_a, reuse_b)
    acc = __builtin_amdgcn_wmma_f32_16x16x4_f32(false, a, false, b,
                                                (short)0, acc, false, false);
  }

  float*      Out  = (which == 0) ? L : R;
  const float bval = (which == 0) ? bias[ncol] : 0.0f;  // fold bias into L
  #pragma unroll
  for (int r = 0; r < 8; ++r) {
    const int m = m0 + r + (half << 3);   // D layout: VGPR r -> M=r / M=8+r
    Out[m * FDIM + ncol] = acc[r] + bval;
  }
}

// ---------------------------------------------------------------------------
// Stage 2: out[b,n,m,c,k] = L[b,n,c,k] + R[b,m,c,k]   (write-BW bound)
// One float4 per thread; NT stores keep the 201MB stream out of L2 so the
// hot 3MB L/R working set stays cached.
// ---------------------------------------------------------------------------
__global__ void __launch_bounds__(256)
bcast_add_kernel(const float* __restrict__ L,
                 const float* __restrict__ R,
                 float* __restrict__ out) {
  const unsigned i = blockIdx.x * 256u + threadIdx.x;   // float4 index
  // decompose i -> (b, n, m, c, k4): k4 in [0,32), c in [0,3), m,n in [0,128)
  const unsigned k4 = i & 31u;
  unsigned t = i >> 5;
  const unsigned c = t % 3u;
  t /= 3u;
  const unsigned m = t & 127u;
  t >>= 7;
  const unsigned n = t & 127u;
  const unsigned b = t >> 7;

  const v4f* __restrict__ L4 = (const v4f*)L;
  const v4f* __restrict__ R4 = (const v4f*)R;
  const v4f l = L4[((b * 128u + n) * 3u + c) * 32u + k4];
  const v4f r = R4[((b * 128u + m) * 3u + c) * 32u + k4];
  __builtin_nontemporal_store(l + r, ((v4f*)out) + i);
}

extern "C" void kernel_launch(void* const* d_in, const int* in_sizes, int n_in,
                              void* d_out, int out_size, void* d_ws, size_t ws_size,
                              hipStream_t stream) {
  const float* X    = (const float*)d_in[0];   // (8,128,3,128) = (3072,128)
  const float* W    = (const float*)d_in[1];   // (256,128)
  const float* bias = (const float*)d_in[2];   // (1,128)
  float*       out  = (float*)d_out;           // (8,128,128,3,128)

  float* L = (float*)d_ws;                     // 3072*128 f32 = 1.5 MB
  float* R = L + ROWS * FDIM;                  // another 1.5 MB

  // Stage 1: 2 * 192 * 8 = 3072 tiles; 1 wave/tile; 4 waves/block -> 768 blocks
  gemm_lr_kernel<<<768, 128, 0, stream>>>(X, W, bias, L, R);

  // Stage 2: 50331648 f32 / 4 = 12582912 float4s; 256 threads/block
  bcast_add_kernel<<<12582912 / 256, 256, 0, stream>>>(L, R, out);
}